// KernelExpansion_68564857913960
// MI455X (gfx1250) — compile-verified
//
#include <hip/hip_runtime.h>
#include <hip/hip_bf16.h>

typedef __attribute__((ext_vector_type(2))) float v2f;
typedef __attribute__((ext_vector_type(8))) float v8f;

#define DDIM 24
#define LOG2E 1.44269504088896340736f

// --- prep: ssqc[n] = coef * ||s_n||^2  (coef = -0.5*log2e/len^2),
//           wp[n]   = w[n] * sigma^2 ------------------------------------
__global__ void kexp_prep_s(const float* __restrict__ samples,
                            const float* __restrict__ weights,
                            const float* __restrict__ sigma_p,
                            const float* __restrict__ len_p,
                            float* __restrict__ ssqc, float* __restrict__ wp,
                            int N) {
    int n = blockIdx.x * blockDim.x + threadIdx.x;
    if (n >= N) return;
    float s = 0.0f;
#pragma unroll
    for (int d = 0; d < DDIM; ++d) {
        float v = samples[n * DDIM + d];
        s = fmaf(v, v, s);
    }
    const float len  = len_p[0];
    const float coef = (-0.5f * LOG2E) / (len * len);
    ssqc[n] = coef * s;
    const float sg = sigma_p[0];
    wp[n] = weights[n] * sg * sg;
}

// --- prep: xh[m] = -0.5 * ||x_m||^2  (WMMA C-seed term) ------------------
__global__ void kexp_prep_x(const float* __restrict__ x,
                            float* __restrict__ xh, int M) {
    int m = blockIdx.x * blockDim.x + threadIdx.x;
    if (m >= M) return;
    float s = 0.0f;
#pragma unroll
    for (int d = 0; d < DDIM; ++d) {
        float v = x[m * DDIM + d];
        s = fmaf(v, v, s);
    }
    xh[m] = -0.5f * s;
}

// --- main: per wave, one 16-wide m tile; loop N in 16-row WMMA tiles -----
__global__ __launch_bounds__(256) void kexp_main(
    const float* __restrict__ x, const float* __restrict__ samples,
    const float* __restrict__ ssqc, const float* __restrict__ wp,
    const float* __restrict__ xh, const float* __restrict__ len_p,
    float* __restrict__ out, int N, int M) {
    const int lane = threadIdx.x & 31;
    const int wave = threadIdx.x >> 5;
    const int lo   = lane & 15;
    const int hi   = lane >> 4;  // 0 or 1
    const int m_base = (blockIdx.x * 8 + wave) * 16;
    if (m_base >= M) return;   // wave-uniform: EXEC stays all-ones for WMMA

    const float len = len_p[0];
    const float c2  = LOG2E / (len * len);   // -2*coef  (coef < 0)

    const int m = m_base + lo;

    // B matrix (x tile), loop-invariant: B[j] holds K = 4j+2*hi, 4j+2*hi+1
    v2f B[6];
#pragma unroll
    for (int j = 0; j < 6; ++j) {
        B[j] = *reinterpret_cast<const v2f*>(x + m * DDIM + 4 * j + 2 * hi);
    }

    // Seed the accumulator with -0.5*||x_m||^2 so the m-term rides the WMMA:
    // D[v] = cross[n,m] - 0.5*||x_m||^2 ;  arg = fma(c2, D, ssqc[n]).
    const float xhv = xh[m];
    v8f cinit;
#pragma unroll
    for (int v = 0; v < 8; ++v) cinit[v] = xhv;

    float acc[8];
#pragma unroll
    for (int v = 0; v < 8; ++v) acc[v] = 0.0f;

    for (int nb = 0; nb < N; nb += 16) {   // N assumed multiple of 16 (8192)
        const int row = nb + lo;
        v2f A[6];
#pragma unroll
        for (int j = 0; j < 6; ++j) {
            A[j] = *reinterpret_cast<const v2f*>(samples + row * DDIM + 4 * j + 2 * hi);
        }
        v8f c = __builtin_amdgcn_wmma_f32_16x16x4_f32(
                    false, A[0], false, B[0], (short)0, cinit, false, false);
#pragma unroll
        for (int j = 1; j < 6; ++j) {
            c = __builtin_amdgcn_wmma_f32_16x16x4_f32(
                    false, A[j], false, B[j], (short)0, c, false, false);
        }

        // C layout: VGPR v holds row n = nb + v + hi*8, col m = m_base + lo
        const float4 s0 = *reinterpret_cast<const float4*>(ssqc + nb + hi * 8);
        const float4 s1 = *reinterpret_cast<const float4*>(ssqc + nb + hi * 8 + 4);
        const float4 w0 = *reinterpret_cast<const float4*>(wp   + nb + hi * 8);
        const float4 w1 = *reinterpret_cast<const float4*>(wp   + nb + hi * 8 + 4);
        const float sv[8] = {s0.x, s0.y, s0.z, s0.w, s1.x, s1.y, s1.z, s1.w};
        const float wv[8] = {w0.x, w0.y, w0.z, w0.w, w1.x, w1.y, w1.z, w1.w};
#pragma unroll
        for (int v = 0; v < 8; ++v) {
            float a = fmaf(c2, c[v], sv[v]);   // coef * sqdist (<= 0 in exact math)
            a = fminf(a, 0.0f);                // clamp sqdist >= 0
            acc[v] = fmaf(wv[v], __builtin_amdgcn_exp2f(a), acc[v]);
        }
    }

    // tree-reduce the 8 row accumulators, then combine lane halves:
    // lanes 0-15 hold rows 0-7, lanes 16-31 rows 8-15 (same m) -> SWAPX16.
    float t0 = (acc[0] + acc[1]) + (acc[2] + acc[3]);
    float t1 = (acc[4] + acc[5]) + (acc[6] + acc[7]);
    float accs = t0 + t1;
    int swz = __builtin_amdgcn_ds_swizzle(__float_as_int(accs), 0x401F);
    float tot = accs + __int_as_float(swz);
    if (lane < 16) out[m] = tot;
}

extern "C" void kernel_launch(void* const* d_in, const int* in_sizes, int n_in,
                              void* d_out, int out_size, void* d_ws, size_t ws_size,
                              hipStream_t stream) {
    (void)n_in; (void)out_size; (void)ws_size;
    const float* x       = (const float*)d_in[0];  // [M, 24]
    const float* samples = (const float*)d_in[1];  // [N, 24]
    const float* weights = (const float*)d_in[2];  // [N]
    const float* sigma   = (const float*)d_in[3];  // [1]
    const float* length  = (const float*)d_in[4];  // [1]
    float* out = (float*)d_out;                    // [M]

    const int M = in_sizes[0] / DDIM;
    const int N = in_sizes[2];

    float* ws   = (float*)d_ws;
    float* ssqc = ws;             // N floats  (coef * ||s||^2)
    float* wp   = ws + N;         // N floats  (w * sigma^2)
    float* xh   = ws + 2 * N;     // M floats  (-0.5 * ||x||^2)

    kexp_prep_s<<<(N + 255) / 256, 256, 0, stream>>>(samples, weights, sigma,
                                                     length, ssqc, wp, N);
    kexp_prep_x<<<(M + 255) / 256, 256, 0, stream>>>(x, xh, M);

    // 8 waves/block, each wave owns a 16-wide m tile -> 128 m per block
    dim3 grid((M + 127) / 128);
    kexp_main<<<grid, 256, 0, stream>>>(x, samples, ssqc, wp, xh, length, out, N, M);
}